// SeqCondAttention_39874476376260
// MI455X (gfx1250) — compile-verified
//
#include <hip/hip_runtime.h>
#include <math.h>

#define D_MODEL 1024
#define K_HEADS 32
#define ANCHOR  4
#define M_THETA 4
#define CONV_K  4
#define H_DIM   32
#define TOTAL   2080
#define HM      128          // H_DIM * M_THETA
#define BATCH   4
#define SEQ     2048
#define MROWS   (BATCH*SEQ)  // 8192

typedef __attribute__((ext_vector_type(2))) float v2f;
typedef __attribute__((ext_vector_type(8))) float v8f;

// ---- CDNA5 async global->LDS path (ASYNCcnt), guarded so fallback compiles ----
#if defined(__has_builtin)
#  if __has_builtin(__builtin_amdgcn_global_load_async_to_lds_b128) && \
      __has_builtin(__builtin_amdgcn_s_wait_asynccnt)
#    define USE_ASYNC_LDS 1
#  endif
#endif
#ifndef USE_ASYNC_LDS
#  define USE_ASYNC_LDS 0
#endif

#if USE_ASYNC_LDS
// builtin expects: ( int4 addrspace(1)* src, int4 addrspace(3)* dst, imm off, imm cpol )
typedef int v4i_t __attribute__((vector_size(16)));
typedef __attribute__((address_space(1))) v4i_t* g_b128_t;
typedef __attribute__((address_space(3))) v4i_t* l_b128_t;
#endif

__device__ __forceinline__ void wait_async_lds() {
#if USE_ASYNC_LDS
    __builtin_amdgcn_s_wait_asynccnt(0);   // s_wait_asynccnt 0
#endif
}

// ---------------------------------------------------------------------------
// fp32 WMMA GEMM:  C[M,N] = A[M,K] * B[K,N], row-major. M%128==0, K%32==0,
// N ragged (guarded). Block tile 128(M) x 64(N); 8 waves as 4x2, each wave a
// 32x32 tile = 2x2 v_wmma_f32_16x16x4_f32 accumulators (4 WMMAs per fragment
// quad -> 2 LDS dwords per WMMA). K staged in chunks of 32 through
// double-buffered LDS via async-to-LDS DMA (fallback: b128 loads/ds stores).
// LDS pads: A rows 36 floats, B rows 72 floats -> 16B-aligned float4 stores
// and bank-conflict-free WMMA fragment reads.
// ---------------------------------------------------------------------------
__global__ __launch_bounds__(256)
void gemm_f32_wmma(const float* __restrict__ A, const float* __restrict__ B,
                   float* __restrict__ C, int N, int Kd,
                   int lda, int ldb, int ldc)
{
    __shared__ float sA[2][128][36];   // 36 KB
    __shared__ float sB[2][32][72];    // 18 KB

    const int tid   = threadIdx.x;
    const int lane  = tid & 31;
    const int wid   = tid >> 5;
    const int waveM = (wid >> 1) * 32;   // 0,32,64,96
    const int waveN = (wid & 1) * 32;    // 0,32
    const int blockN = blockIdx.x * 64;
    const int blockM = blockIdx.y * 128;

    // ISA 7.12.2 f32 fragments: lanes 0-15 -> K pair {0,1}, lanes 16-31 -> {2,3}
    const int mfrag = lane & 15;
    const int kfrag = (lane >> 4) * 2;

    auto stage = [&](int buf, int k0) {
        // ---- A tile 128x32: 4 float4 per thread ----
        #pragma unroll
        for (int i = 0; i < 4; ++i) {
            const int t  = tid + i * 256;
            const int r  = t >> 3;
            const int c4 = (t & 7) * 4;
            const float* gp = A + (size_t)(blockM + r) * lda + k0 + c4;
#if USE_ASYNC_LDS
            __builtin_amdgcn_global_load_async_to_lds_b128(
                (g_b128_t)gp, (l_b128_t)&sA[buf][r][c4], 0, 0);
#else
            const float4 v = *(const float4*)gp;
            *(float4*)&sA[buf][r][c4] = v;
#endif
        }
        // ---- B tile 32x64: 2 float4 per thread (guard ragged N) ----
        const bool fullN = (blockN + 64 <= N);
        #pragma unroll
        for (int i = 0; i < 2; ++i) {
            const int t  = tid + i * 256;
            const int r  = t >> 4;
            const int c4 = (t & 15) * 4;
            const float* gp = B + (size_t)(k0 + r) * ldb + blockN + c4;
            if (fullN) {
#if USE_ASYNC_LDS
                __builtin_amdgcn_global_load_async_to_lds_b128(
                    (g_b128_t)gp, (l_b128_t)&sB[buf][r][c4], 0, 0);
#else
                const float4 v = *(const float4*)gp;
                *(float4*)&sB[buf][r][c4] = v;
#endif
            } else {
                #pragma unroll
                for (int e = 0; e < 4; ++e) {
                    const int col = blockN + c4 + e;
                    sB[buf][r][c4 + e] = (col < N) ? gp[e] : 0.0f;
                }
            }
        }
    };

    v8f acc[2][2] = {{{}, {}}, {{}, {}}};

    const int nchunks = Kd >> 5;
    stage(0, 0);
    wait_async_lds();
    __syncthreads();

    for (int ch = 0; ch < nchunks; ++ch) {
        const int cur = ch & 1;
        if (ch + 1 < nchunks) stage(cur ^ 1, (ch + 1) << 5);   // overlap DMA w/ math

        #pragma unroll
        for (int kk = 0; kk < 32; kk += 4) {
            v2f a0, a1, b0, b1;
            a0.x = sA[cur][waveM +      mfrag][kk + kfrag];
            a0.y = sA[cur][waveM +      mfrag][kk + kfrag + 1];
            a1.x = sA[cur][waveM + 16 + mfrag][kk + kfrag];
            a1.y = sA[cur][waveM + 16 + mfrag][kk + kfrag + 1];
            b0.x = sB[cur][kk + kfrag    ][waveN +      mfrag];
            b0.y = sB[cur][kk + kfrag + 1][waveN +      mfrag];
            b1.x = sB[cur][kk + kfrag    ][waveN + 16 + mfrag];
            b1.y = sB[cur][kk + kfrag + 1][waveN + 16 + mfrag];
            acc[0][0] = __builtin_amdgcn_wmma_f32_16x16x4_f32(false, a0, false, b0, (short)0, acc[0][0], false, false);
            acc[0][1] = __builtin_amdgcn_wmma_f32_16x16x4_f32(false, a0, false, b1, (short)0, acc[0][1], false, false);
            acc[1][0] = __builtin_amdgcn_wmma_f32_16x16x4_f32(false, a1, false, b0, (short)0, acc[1][0], false, false);
            acc[1][1] = __builtin_amdgcn_wmma_f32_16x16x4_f32(false, a1, false, b1, (short)0, acc[1][1], false, false);
        }
        wait_async_lds();
        __syncthreads();
    }

    // C/D layout: VGPR r -> lanes 0-15: row r, lanes 16-31: row r+8; col = lane&15
    #pragma unroll
    for (int mi = 0; mi < 2; ++mi) {
        #pragma unroll
        for (int ni = 0; ni < 2; ++ni) {
            const int rowBase = blockM + waveM + mi * 16 + (lane >> 4) * 8;
            const int col     = blockN + waveN + ni * 16 + (lane & 15);
            if (col < N) {
                #pragma unroll
                for (int r = 0; r < 8; ++r)
                    C[(size_t)(rowBase + r) * ldc + col] = acc[mi][ni][r];
            }
        }
    }
}

// ---------------------------------------------------------------------------
// Causal depthwise conv1d (k=4, left zero pad) + split into x_val / gate / s_raw
// ---------------------------------------------------------------------------
__global__ void conv_split(const float* __restrict__ z, const float* __restrict__ wconv,
                           const float* __restrict__ bconv,
                           float* __restrict__ xv, float* __restrict__ gate,
                           float* __restrict__ sr)
{
    size_t idx = (size_t)blockIdx.x * blockDim.x + threadIdx.x;
    const size_t n = (size_t)BATCH * SEQ * TOTAL;
    if (idx >= n) return;
    const int    c  = (int)(idx % TOTAL);
    const size_t bl = idx / TOTAL;
    const int    l  = (int)(bl % SEQ);

    float acc = bconv[c];
    #pragma unroll
    for (int j = 0; j < CONV_K; ++j) {
        int ll = l - (CONV_K - 1) + j;
        if (ll >= 0)
            acc += z[(bl + (size_t)(ll - l)) * TOTAL + c] * wconv[j * TOTAL + c];
    }
    if (c < D_MODEL) {
        xv[bl * D_MODEL + c] = acc;
    } else if (c < 2 * D_MODEL) {
        float g = acc / (1.0f + __expf(-acc));          // silu
        gate[bl * D_MODEL + (c - D_MODEL)] = g;
    } else {
        sr[bl * K_HEADS + (c - 2 * D_MODEL)] = acc;
    }
}

// ---------------------------------------------------------------------------
// Fused: phase features -> cumulative scan over L (den collapsed: p_w is
// broadcast over (H,M), so 1 den chain/head instead of 128) -> 1/den ->
// joint RMS norm -> per-head projection (re_n@W_re + im_n@W_im) -> * gate.
// One workgroup per (b,k): grid 128, block 128 (thread = h*4+m).
// ---------------------------------------------------------------------------
__global__ __launch_bounds__(128)
void scan_heads(const float* __restrict__ xv, const float* __restrict__ sr,
                const float* __restrict__ gate, const float* __restrict__ theta,
                const float* __restrict__ dec_slopes, const float* __restrict__ anc_slopes,
                const float* __restrict__ score_scale, const float* __restrict__ deriv_logits,
                const float* __restrict__ norm_scale,
                const float* __restrict__ W_re, const float* __restrict__ W_im,
                float* __restrict__ yg)
{
    __shared__ float lWre[HM * H_DIM];   // 16 KB
    __shared__ float lWim[HM * H_DIM];   // 16 KB
    __shared__ float sRe[HM], sIm[HM];
    __shared__ float wsum[4];
    __shared__ float proj[128];

    const int tid = threadIdx.x;
    const int b   = blockIdx.x / K_HEADS;
    const int k   = blockIdx.x % K_HEADS;
    const int h   = tid >> 2;
    const int m   = tid & 3;

    for (int i = tid; i < HM * H_DIM; i += 128) {
        lWre[i] = W_re[i];
        lWim[i] = W_im[i];
    }

    // per-head parameters
    const bool  is_dec = (k < K_HEADS - ANCHOR);
    const float raw    = is_dec ? dec_slopes[k] : anc_slopes[k - (K_HEADS - ANCHOR)];
    const float slope  = log1pf(__expf(raw));                 // softplus
    const float ss     = score_scale[k];
    const float th     = theta[((k * H_DIM) + h) * M_THETA + m];
    // softmax(deriv_logits)
    const float d0 = deriv_logits[0], d1 = deriv_logits[1], d2 = deriv_logits[2];
    const float dm = fmaxf(d0, fmaxf(d1, d2));
    const float e0 = __expf(d0 - dm), e1 = __expf(d1 - dm), e2 = __expf(d2 - dm);
    const float esi = 1.0f / (e0 + e1 + e2);
    const float w0 = e0 * esi, w1 = e1 * esi, w2 = e2 * esi;
    const float nsr = norm_scale[tid], nsi = norm_scale[HM + tid];

    __syncthreads();

    float accRe = 0.f, accIm = 0.f, accD = 0.f;
    const int    q      = tid >> 5;     // projection K-slice
    const int    hh     = tid & 31;     // projection output column
    const size_t baseBL = (size_t)b * SEQ;

    for (int l = 0; l < SEQ; ++l) {
        const size_t bl = baseBL + l;
        const float x = xv[bl * D_MODEL + k * H_DIM + h];
        const float s = sr[bl * K_HEADS + k];
        const float p = __expf(fminf(fmaxf(ss * s, -20.f), 20.f));
        const float dist = is_dec ? (float)(SEQ - 1 - l) : (float)l;
        const float pw = p * __expf(-slope * dist);
        float sphi, cphi;
        __sincosf(x * th, &sphi, &cphi);
        const float poly = w0 + w1 * x - w2 * x * x;
        accRe = fmaf(pw, poly * cphi, accRe);
        accIm = fmaf(pw, poly * sphi, accIm);
        accD += pw;                                  // identical across threads

        const float inv = 1.0f / fmaxf(accD, 1e-4f);
        const float re = accRe * inv, im = accIm * inv;

        // joint RMS over 2*HM values: wave reduce, then 4-wave combine
        float part = re * re + im * im;
        #pragma unroll
        for (int o = 16; o >= 1; o >>= 1) part += __shfl_xor(part, o, 32);
        if ((tid & 31) == 0) wsum[tid >> 5] = part;
        __syncthreads();
        const float ms = (wsum[0] + wsum[1] + wsum[2] + wsum[3]) * (1.0f / (2.0f * HM));
        const float rq = rsqrtf(ms + 1e-5f);
        sRe[tid] = re * rq * nsr;
        sIm[tid] = im * rq * nsi;
        __syncthreads();

        // projection: y[hh] = sum_j reN[j]*W_re[j,hh] + imN[j]*W_im[j,hh]
        float accP = 0.f;
        #pragma unroll
        for (int j = 0; j < 32; ++j) {
            const int jj = q * 32 + j;
            accP = fmaf(sRe[jj], lWre[jj * H_DIM + hh], accP);
            accP = fmaf(sIm[jj], lWim[jj * H_DIM + hh], accP);
        }
        proj[tid] = accP;
        __syncthreads();
        if (tid < 32) {
            const float y = proj[tid] + proj[32 + tid] + proj[64 + tid] + proj[96 + tid];
            const float g = gate[bl * D_MODEL + k * H_DIM + tid];
            yg[bl * D_MODEL + k * H_DIM + tid] = y * g;
        }
        __syncthreads();   // protect wsum/sRe/proj reuse next iteration
    }
}

// ---------------------------------------------------------------------------
extern "C" void kernel_launch(void* const* d_in, const int* in_sizes, int n_in,
                              void* d_out, int out_size, void* d_ws, size_t ws_size,
                              hipStream_t stream) {
    (void)in_sizes; (void)n_in; (void)out_size; (void)ws_size;
    const float* x     = (const float*)d_in[0];
    const float* W_in  = (const float*)d_in[1];
    const float* wconv = (const float*)d_in[2];
    const float* bconv = (const float*)d_in[3];
    const float* theta = (const float*)d_in[4];
    const float* decs  = (const float*)d_in[5];
    const float* ancs  = (const float*)d_in[6];
    const float* sscal = (const float*)d_in[7];
    const float* dlog  = (const float*)d_in[8];
    const float* nscal = (const float*)d_in[9];
    const float* W_re  = (const float*)d_in[10];
    const float* W_im  = (const float*)d_in[11];
    const float* W_out = (const float*)d_in[12];
    float* out = (float*)d_out;

    // workspace carve-out (fp32): z(8192x2080) | xv(8192x1024) | gate | sr(8192x32) | yg
    float* z    = (float*)d_ws;
    float* xv   = z    + (size_t)MROWS * TOTAL;
    float* gate = xv   + (size_t)MROWS * D_MODEL;
    float* sr   = gate + (size_t)MROWS * D_MODEL;
    float* yg   = sr   + (size_t)MROWS * K_HEADS;

    // 1) z = x @ W_in   (8192x1024)(1024x2080)
    dim3 g1((TOTAL + 63) / 64, MROWS / 128);
    gemm_f32_wmma<<<g1, 256, 0, stream>>>(x, W_in, z, TOTAL, D_MODEL, D_MODEL, TOTAL, TOTAL);

    // 2) causal depthwise conv + split/activations
    const size_t nconv = (size_t)BATCH * SEQ * TOTAL;
    conv_split<<<(unsigned)((nconv + 255) / 256), 256, 0, stream>>>(z, wconv, bconv, xv, gate, sr);

    // 3) fused scan + norm + head projection + gating
    scan_heads<<<BATCH * K_HEADS, 128, 0, stream>>>(xv, sr, gate, theta, decs, ancs,
                                                    sscal, dlog, nscal, W_re, W_im, yg);

    // 4) out = yg @ W_out   (8192x1024)(1024x1024)
    dim3 g2(D_MODEL / 64, MROWS / 128);
    gemm_f32_wmma<<<g2, 256, 0, stream>>>(yg, W_out, out, D_MODEL, D_MODEL, D_MODEL, D_MODEL, D_MODEL);
}